// SimpleHeterogeneousMicrobialGNN_59502476919387
// MI455X (gfx1250) — compile-verified
//
#include <hip/hip_runtime.h>
#include <math.h>

#define N_NODES 100000
#define N_EDGES 1600000
#define H 32
#define R 3
#define LN_EPS 1e-5f

typedef float v2f __attribute__((ext_vector_type(2)));
typedef float v8f __attribute__((ext_vector_type(8)));

// D = A(16x4 f32) * B(4x16 f32) + C(16x16 f32), full fp32 WMMA on CDNA5.
__device__ __forceinline__ v8f wmma_f32_16x16x4(v2f a, v2f b, v8f c) {
    return __builtin_amdgcn_wmma_f32_16x16x4_f32(false, a, false, b, (short)0, c, false, false);
}

// -------------------------------------------------------------------------
// h[n,j] = x[n] * W_emb[j] + b_emb[j]
__global__ void embed_kernel(const float* __restrict__ x, const float* __restrict__ Wemb,
                             const float* __restrict__ bemb, float* __restrict__ h) {
    int t = blockIdx.x * blockDim.x + threadIdx.x;
    if (t >= N_NODES * H) return;
    int node = t >> 5, j = t & 31;
    h[t] = x[node] * Wemb[j] + bemb[j];
}

// -------------------------------------------------------------------------
// Per-edge relation prediction. One wave handles 16 edges.
// Layer1 (64->16) via 16 chained f32 WMMAs, A-fragment gathered in-register.
__global__ void edge_type_kernel(const float* __restrict__ h, const int* __restrict__ ei,
                                 const float* __restrict__ w1, const float* __restrict__ b1,
                                 const float* __restrict__ w2, const float* __restrict__ b2,
                                 unsigned char* __restrict__ etype) {
    __shared__ float lw1[64 * 16];      // W1^T : [k][n]
    __shared__ float lw2[R * 16];
    __shared__ float lb1[16];
    __shared__ float lb2[R];
    __shared__ float lhid[8][16 * 17];  // per-wave hidden staging (padded)

    for (int i = threadIdx.x; i < 64 * 16; i += blockDim.x) {
        int k = i >> 4, n = i & 15;
        lw1[i] = w1[n * 64 + k];
    }
    if (threadIdx.x < R * 16) lw2[threadIdx.x] = w2[threadIdx.x];
    if (threadIdx.x < 16)     lb1[threadIdx.x] = b1[threadIdx.x];
    if (threadIdx.x < R)      lb2[threadIdx.x] = b2[threadIdx.x];
    __syncthreads();

    int wid  = threadIdx.x >> 5;
    int lane = threadIdx.x & 31;
    int tile = blockIdx.x * (blockDim.x >> 5) + wid;
    if (tile * 16 >= N_EDGES) return;

    int m    = lane & 15;     // edge within tile (A rows) == hidden col (B cols)
    int half = lane >> 4;
    int kb   = half * 2;      // WMMA 16x4 f32 layout: lane-half selects K pair
    int e    = tile * 16 + m;
    int row  = ei[e];
    int col  = ei[N_EDGES + e];

    float bv = lb1[m];
    v8f acc = {bv, bv, bv, bv, bv, bv, bv, bv};
    #pragma unroll
    for (int ks = 0; ks < 16; ++ks) {
        int kk = ks * 4 + kb;                 // feature index in concat [h[row]|h[col]]
        int f  = kk & 31;
        const float* src = (kk < 32) ? (h + (size_t)row * H + f)
                                     : (h + (size_t)col * H + f);
        v2f a = *(const v2f*)src;             // 8B-aligned (kk even)
        v2f b; b.x = lw1[kk * 16 + m]; b.y = lw1[(kk + 1) * 16 + m];
        acc = wmma_f32_16x16x4(a, b, acc);
    }
    #pragma unroll
    for (int v = 0; v < 8; ++v)               // C layout: row = v + half*8, col = m
        lhid[wid][(v + half * 8) * 17 + m] = fmaxf(acc[v], 0.0f);
    asm volatile("s_wait_dscnt 0x0" ::: "memory");   // wave-local cross-lane LDS sync

    if (half == 0) {                          // lanes 0..15: edge = lane
        float best = -1e30f; int bi = 0;
        #pragma unroll
        for (int r = 0; r < R; ++r) {
            float s = lb2[r];
            #pragma unroll
            for (int j = 0; j < 16; ++j) s += lhid[wid][lane * 17 + j] * lw2[r * 16 + j];
            if (s > best) { best = s; bi = r; }   // strict > : first-max like argmax
        }
        etype[e] = (unsigned char)bi;
    }
}

// -------------------------------------------------------------------------
__global__ void zero_kernel(float* __restrict__ p, long long n) {
    long long t = (long long)blockIdx.x * blockDim.x + threadIdx.x;
    if (t < n) p[t] = 0.0f;
}

// One wave per edge; lane = feature. Coalesced gather + L2-resident atomics.
__global__ void scatter_kernel(const float* __restrict__ h, const int* __restrict__ ei,
                               const unsigned char* __restrict__ etype,
                               float* __restrict__ num, float* __restrict__ cnt) {
    long long t = (long long)blockIdx.x * blockDim.x + threadIdx.x;
    long long e = t >> 5;
    int j = (int)(t & 31);
    if (e >= N_EDGES) return;
    int row = ei[e];
    int col = ei[N_EDGES + e];
    int r   = etype[e];
    float val = h[(size_t)row * H + j];
    atomicAdd(&num[((size_t)r * N_NODES + col) * H + j], val);
    if (j == 0) atomicAdd(&cnt[(size_t)r * N_NODES + col], 1.0f);
}

// -------------------------------------------------------------------------
// out = h@root + bias + sum_r mean_r@W_r ; h' = relu(LN(out)) + h
// One wave per 16-node tile; 64 f32 WMMAs per tile.
__global__ void node_update_kernel(const float* __restrict__ hin, float* __restrict__ hout,
                                   const float* __restrict__ num, const float* __restrict__ cnt,
                                   const float* __restrict__ root, const float* __restrict__ wr,
                                   const float* __restrict__ bias, const float* __restrict__ lng,
                                   const float* __restrict__ lnb) {
    __shared__ float lw[(1 + R) * H * H];     // [root | W_0 | W_1 | W_2], row-major [k][n]
    __shared__ float stag[8][16 * 33];        // per-wave 16x32 out tile, stride 33 (bank-safe)
    for (int i = threadIdx.x; i < (1 + R) * H * H; i += blockDim.x)
        lw[i] = (i < H * H) ? root[i] : wr[i - H * H];
    __syncthreads();

    int wid  = threadIdx.x >> 5;
    int lane = threadIdx.x & 31;
    int tile = blockIdx.x * (blockDim.x >> 5) + wid;
    if (tile * 16 >= N_NODES) return;         // wave-uniform guard

    int m    = lane & 15;
    int half = lane >> 4;
    int kb   = half * 2;
    int node = tile * 16 + m;
    const float* hrow = hin + (size_t)node * H;

    float rc[R];
    #pragma unroll
    for (int r = 0; r < R; ++r)
        rc[r] = 1.0f / fmaxf(cnt[(size_t)r * N_NODES + node], 1.0f);

    #pragma unroll
    for (int ct = 0; ct < 2; ++ct) {          // two 16-wide column tiles of the 32-wide output
        int n = ct * 16 + m;                  // output column for B/C role of this lane
        float bv = bias[n];
        v8f acc = {bv, bv, bv, bv, bv, bv, bv, bv};
        #pragma unroll
        for (int ks = 0; ks < 8; ++ks) {      // h @ root
            int kk = ks * 4 + kb;
            v2f a = *(const v2f*)(hrow + kk);
            v2f b; b.x = lw[kk * H + n]; b.y = lw[(kk + 1) * H + n];
            acc = wmma_f32_16x16x4(a, b, acc);
        }
        #pragma unroll
        for (int r = 0; r < R; ++r) {         // mean_r @ W_r (mean fused into A load)
            const float* nrow = num + ((size_t)r * N_NODES + node) * H;
            const float* wb   = lw + (1 + r) * H * H;
            float sc = rc[r];
            #pragma unroll
            for (int ks = 0; ks < 8; ++ks) {
                int kk = ks * 4 + kb;
                v2f a; a.x = nrow[kk] * sc; a.y = nrow[kk + 1] * sc;
                v2f b; b.x = wb[kk * H + n]; b.y = wb[(kk + 1) * H + n];
                acc = wmma_f32_16x16x4(a, b, acc);
            }
        }
        #pragma unroll
        for (int v = 0; v < 8; ++v)
            stag[wid][(v + half * 8) * 33 + n] = acc[v];
    }
    asm volatile("s_wait_dscnt 0x0" ::: "memory");

    // LayerNorm over H=32 per row; both lane-halves compute stats for row m,
    // each half writes its 16 columns. relu + residual.
    float mu = 0.0f;
    #pragma unroll
    for (int j = 0; j < H; ++j) mu += stag[wid][m * 33 + j];
    mu *= (1.0f / H);
    float var = 0.0f;
    #pragma unroll
    for (int j = 0; j < H; ++j) { float d = stag[wid][m * 33 + j] - mu; var += d * d; }
    var *= (1.0f / H);
    float rsig = rsqrtf(var + LN_EPS);
    #pragma unroll
    for (int jj = 0; jj < 16; ++jj) {
        int j = half * 16 + jj;
        float v = (stag[wid][m * 33 + j] - mu) * rsig * lng[j] + lnb[j];
        hout[(size_t)node * H + j] = fmaxf(v, 0.0f) + hrow[j];
    }
}

// -------------------------------------------------------------------------
extern "C" void kernel_launch(void* const* d_in, const int* in_sizes, int n_in,
                              void* d_out, int out_size, void* d_ws, size_t ws_size,
                              hipStream_t stream) {
    const float* x      = (const float*)d_in[0];
    const int*   ei     = (const int*)d_in[1];     // [2,E] int32 per harness ABI
    const float* Wemb   = (const float*)d_in[2];
    const float* bemb   = (const float*)d_in[3];
    const float* rpw1   = (const float*)d_in[4];
    const float* rpb1   = (const float*)d_in[5];
    const float* rpw2   = (const float*)d_in[6];
    const float* rpb2   = (const float*)d_in[7];
    const float* rgw    = (const float*)d_in[8];   // [L,R,H,H]
    const float* rgroot = (const float*)d_in[9];   // [L,H,H]
    const float* rgbias = (const float*)d_in[10];  // [L,H]
    const float* lng    = (const float*)d_in[11];  // [L,H]
    const float* lnb    = (const float*)d_in[12];  // [L,H]
    float* out = (float*)d_out;

    // Workspace layout (floats): h0 | h1 | num[R,N,H] | cnt[R,N] | etype(bytes)
    float* ws  = (float*)d_ws;
    float* h0  = ws;
    float* h1  = h0 + (size_t)N_NODES * H;
    float* num = h1 + (size_t)N_NODES * H;
    float* cnt = num + (size_t)R * N_NODES * H;
    unsigned char* etype = (unsigned char*)(cnt + (size_t)R * N_NODES);

    { // embedding
        int total = N_NODES * H;
        embed_kernel<<<(total + 255) / 256, 256, 0, stream>>>(x, Wemb, bemb, h0);
    }
    { // edge types (once, from initial embeddings)
        int tiles  = N_EDGES / 16;       // 100000
        int blocks = (tiles + 7) / 8;    // 8 waves/block
        edge_type_kernel<<<blocks, 256, 0, stream>>>(h0, ei, rpw1, rpb1, rpw2, rpb2, etype);
    }

    long long zn = (long long)R * N_NODES * H + (long long)R * N_NODES; // num+cnt contiguous
    for (int l = 0; l < 2; ++l) {
        const float* hin = (l == 0) ? h0 : h1;
        float*      hout = (l == 0) ? h1 : out;
        zero_kernel<<<(int)((zn + 255) / 256), 256, 0, stream>>>(num, zn);
        {
            long long total = (long long)N_EDGES * 32;  // wave per edge
            scatter_kernel<<<(int)((total + 255) / 256), 256, 0, stream>>>(hin, ei, etype, num, cnt);
        }
        {
            int tiles  = (N_NODES + 15) / 16;  // 6250
            int blocks = (tiles + 7) / 8;
            node_update_kernel<<<blocks, 256, 0, stream>>>(
                hin, hout, num, cnt,
                rgroot + (size_t)l * H * H,
                rgw    + (size_t)l * R * H * H,
                rgbias + (size_t)l * H,
                lng    + (size_t)l * H,
                lnb    + (size_t)l * H);
        }
    }
}